// GraphPoolModule_77678778515952
// MI455X (gfx1250) — compile-verified
//
#include <hip/hip_runtime.h>

// ---------------------------------------------------------------------------
// Graph pooling (gather + ragged segment mean), MI455X / gfx1250.
//   input  [N_IN, 128] f32, idxn [E] i32, degs [S] i32  ->  out [S, 128] f32
// Memory-bound: ~335 MB traffic @ 23.3 TB/s ~= 14 us. No WMMA structure;
// the CDNA5 win is GLOBAL_LOAD_ASYNC_TO_LDS_B128 + ASYNCcnt pipelining of
// the random 512 B row gathers (deep MLP without burning return VGPRs).
// ---------------------------------------------------------------------------

#define NBUF 8          // async ring slots per wave (8 x 512 B = 4 KB LDS)
#define SCAN_T 256

// one async instruction = 32 lanes x 16 B = one contiguous 512 B row
__device__ __forceinline__ void async_row16(unsigned lds_addr,
                                            unsigned long long gaddr) {
  asm volatile("global_load_async_to_lds_b128 %0, %1, off"
               :: "v"(lds_addr), "v"(gaddr) : "memory");
}

// ---------------------------------------------------------------------------
// Kernel 1: exclusive scan of degs -> CSR offsets (single workgroup).
// ---------------------------------------------------------------------------
__global__ __launch_bounds__(SCAN_T) void scan_offsets(
    const int* __restrict__ degs, int S, int* __restrict__ offs) {
  __shared__ int sh[SCAN_T];
  const int t = threadIdx.x;
  const int per = (S + SCAN_T - 1) / SCAN_T;
  const int begin = t * per;
  const int end = (begin + per < S) ? (begin + per) : S;

  int sum = 0;
  for (int i = begin; i < end; ++i) sum += degs[i];
  sh[t] = sum;
  __syncthreads();

  // Hillis-Steele inclusive scan over per-thread chunk totals
  for (int off = 1; off < SCAN_T; off <<= 1) {
    int other = (t >= off) ? sh[t - off] : 0;
    __syncthreads();
    sh[t] += other;
    __syncthreads();
  }

  int run = sh[t] - sum;  // exclusive prefix of this chunk
  for (int i = begin; i < end; ++i) {
    offs[i] = run;
    run += degs[i];
  }
}

// ---------------------------------------------------------------------------
// Kernel 2: one wave32 per segment, C == 128. Lane l owns channels 4l..4l+3.
// 8-deep async global->LDS ring pipelines the 512 B row gathers.
// ---------------------------------------------------------------------------
__global__ __launch_bounds__(32) void pool128_kernel(
    const float* __restrict__ input, const int* __restrict__ idxn,
    const int* __restrict__ offs, const int* __restrict__ degs,
    float* __restrict__ out) {
  __shared__ float4 buf[NBUF * 32];  // 4 KB ring

  const int s = blockIdx.x;
  const int lane = threadIdx.x;
  const int start = offs[s];
  const int deg = degs[s];

  float4 acc = make_float4(0.f, 0.f, 0.f, 0.f);

  if (deg > 0 && deg <= 64) {
    // Preload all segment indices into 2 VGPRs/lane (deg <= 64), clamped.
    const int c0 = (lane < deg) ? lane : (deg - 1);
    const int i0 = idxn[start + c0];
    int i1 = 0;
    if (deg > 32) {
      const int c1 = (32 + lane < deg) ? lane : (deg - 33);
      i1 = idxn[start + 32 + c1];
    }

    const unsigned lds_lane =
        (unsigned)(size_t)(&buf[0]) + (unsigned)(lane * 16);
    const unsigned long long base =
        (unsigned long long)(size_t)input + (unsigned long long)(lane * 16);

    // Prologue: fill the ring.
    const int prol = (deg < NBUF) ? deg : NBUF;
    for (int k = 0; k < prol; ++k) {
      const int row = __shfl((k < 32) ? i0 : i1, k & 31);
      async_row16(lds_lane + (unsigned)((k % NBUF) * 512),
                  base + (unsigned long long)(unsigned)row * 512ull);
    }

    // Steady state: wait-oldest, accumulate, reissue.
    const int main_iters = (deg > NBUF) ? (deg - NBUF) : 0;
    for (int k = 0; k < main_iters; ++k) {
      asm volatile("s_wait_asynccnt 7" ::: "memory");   // slot k written
      const float4 v = buf[(k % NBUF) * 32 + lane];     // ds_load_b128
      acc.x += v.x; acc.y += v.y; acc.z += v.z; acc.w += v.w;
      // WAR hardening: ensure the ring-slot read has fully completed before
      // the async engine may overwrite the same slot (async LDS writes are
      // unordered w.r.t. DS ops per ISA).
      asm volatile("s_wait_dscnt 0" ::: "memory");
      const int kk = k + NBUF;
      const int row = __shfl((kk < 32) ? i0 : i1, kk & 31);
      async_row16(lds_lane + (unsigned)((kk % NBUF) * 512),
                  base + (unsigned long long)(unsigned)row * 512ull);
    }

    // Drain.
    asm volatile("s_wait_asynccnt 0" ::: "memory");
    for (int k = main_iters; k < deg; ++k) {
      const float4 v = buf[(k % NBUF) * 32 + lane];
      acc.x += v.x; acc.y += v.y; acc.z += v.z; acc.w += v.w;
    }
  } else if (deg > 64) {
    // Generic path: uniform scalar index load + direct b128 gather.
    for (int k = 0; k < deg; ++k) {
      const int row = idxn[start + k];
      const float4 v =
          ((const float4*)(input + (size_t)row * 128))[lane];
      acc.x += v.x; acc.y += v.y; acc.z += v.z; acc.w += v.w;
    }
  }

  const float inv = (deg > 0) ? (1.0f / (float)deg) : 0.0f;
  const float4 o = make_float4(acc.x * inv, acc.y * inv,
                               acc.z * inv, acc.w * inv);
  ((float4*)(out + (size_t)s * 128))[lane] = o;
}

// ---------------------------------------------------------------------------
// Fallback for C != 128 (not expected for this harness).
// ---------------------------------------------------------------------------
__global__ void pool_generic(const float* __restrict__ input,
                             const int* __restrict__ idxn,
                             const int* __restrict__ offs,
                             const int* __restrict__ degs,
                             float* __restrict__ out, int C) {
  const int s = blockIdx.x;
  const int start = offs[s];
  const int deg = degs[s];
  for (int c = threadIdx.x; c < C; c += blockDim.x) {
    float acc = 0.f;
    for (int k = 0; k < deg; ++k)
      acc += input[(size_t)idxn[start + k] * C + c];
    out[(size_t)s * C + c] = (deg > 0) ? (acc / (float)deg) : 0.f;
  }
}

// ---------------------------------------------------------------------------
extern "C" void kernel_launch(void* const* d_in, const int* in_sizes, int n_in,
                              void* d_out, int out_size, void* d_ws,
                              size_t ws_size, hipStream_t stream) {
  const float* input = (const float*)d_in[0];
  const int* idxn    = (const int*)d_in[1];
  // d_in[2] = segment_ids (unused: degs + scan reproduce the CSR layout)
  const int* degs    = (const int*)d_in[3];

  const int S = in_sizes[3];
  const int C = (S > 0) ? (out_size / S) : 0;
  int* offs = (int*)d_ws;  // S ints of scratch

  scan_offsets<<<1, SCAN_T, 0, stream>>>(degs, S, offs);

  if (C == 128) {
    pool128_kernel<<<S, 32, 0, stream>>>(input, idxn, offs, degs,
                                         (float*)d_out);
  } else {
    pool_generic<<<S, 256, 0, stream>>>(input, idxn, offs, degs,
                                        (float*)d_out, C);
  }
}